// AutoEncoder_24283745091684
// MI455X (gfx1250) — compile-verified
//
#include <hip/hip_runtime.h>

// ---------------------------------------------------------------------------
// Quantum autoencoder (10-qubit sim + analytic swap test) on MI455X / gfx1250.
//
// Math reduction (exact): wires 10..13 are |0> until the swap test, so
//   out[b] = 0.5 * (1 - p000),  p000 = sum_{idx % 8 == 0} |psi10(idx)|^2
// where psi10 is the 10-qubit state after AngleEmbedding + 4 entangler layers.
//
// Gate fusion: hi group = wires 0..4 (idx>>5), lo group = wires 5..9 (idx&31).
// State = 32x32 complex matrix S[h][l].  Per layer (order-exact, RX commutes
// with CNOT targets):
//     S <- U_hi * S          (U_hi = CNOT-cascade(0..4) * kron RX(0..4))
//     Cross45:  if (h&1) l ^= 16
//     S <- S * U_lo^T        (U_lo likewise on wires 5..9)
//     Cross90:  if (l&1) h ^= 16
// Complex matmul realized as real 64x64 * 64x32 with M = [[Re,-Im],[Im,Re]];
// each result stored complex-transposed so both matmuls are "gate x matrix"
// and both crosses become "swap rows r<->r+16 on odd columns".
//
// Matmuls use V_WMMA_F32_16X16X4_F32 (f32 keeps complex64 accuracy).
// Everything lives in LDS; one workgroup (64 threads, 2 waves) per sample.
// ---------------------------------------------------------------------------

typedef float v2f __attribute__((ext_vector_type(2)));
typedef float v8f __attribute__((ext_vector_type(8)));

#define NQ    10
#define DEPTH 4
#define MS    65   // LDS row stride of 64x64 gate matrices (bank-conflict pad)
#define XS    33   // LDS row stride of 64x32 state buffers  (bank-conflict pad)

__global__ __launch_bounds__(64) void qae_kernel(const float* __restrict__ features,
                                                 const float* __restrict__ weights,
                                                 float* __restrict__ out)
{
    __shared__ float Mhi[64 * MS];        // real embedding of U_hi (64x64)
    __shared__ float Mlo[64 * MS];        // real embedding of U_lo (64x64)
    __shared__ float Xbuf[2][64 * XS];    // ping-pong state, [Re(32 rows); Im(32 rows)] x 32 cols
    __shared__ float fc[NQ], fs[NQ];      // cos/sin(features/2)
    __shared__ float wc[NQ], ws[NQ];      // cos/sin(weights[layer]/2)
    __shared__ float aRe[32], aIm[32], bRe[32], bIm[32];
    __shared__ float red[64];

    const int t    = threadIdx.x;         // 0..63
    const int lane = t & 31;
    const int wave = t >> 5;              // 0..1
    const int samp = blockIdx.x;          // 0..255

    // ---- stage embedding half-angle cos/sin ----
    if (t < NQ) {
        float half = 0.5f * features[samp * NQ + t];
        fc[t] = cosf(half);
        fs[t] = sinf(half);
    }
    __syncthreads();

    // ---- embedding product state: a = kron RX(0..4)|0>, b = kron RX(5..9)|0>.
    // Each set bit contributes (-i)*sin, clear bit cos => value = (-i)^popcount * prod(mags)
    {
        int which = t >> 5, idx = t & 31;
        const float* cc = which ? (fc + 5) : fc;
        const float* ss = which ? (fs + 5) : fs;
        float m = 1.0f;
        #pragma unroll
        for (int j = 0; j < 5; ++j) {
            int bit = (idx >> (4 - j)) & 1;   // wire j of group (wire = MSB of group index)
            m *= bit ? ss[j] : cc[j];
        }
        int k = __popc(idx) & 3;
        float re = (k == 0) ? m : ((k == 2) ? -m : 0.0f);
        float im = (k == 3) ? m : ((k == 1) ? -m : 0.0f);
        if (which) { bRe[idx] = re; bIm[idx] = im; }
        else       { aRe[idx] = re; aIm[idx] = im; }
    }
    __syncthreads();

    // ---- initial state (HI-major): S[h][l] = a[h]*b[l] ----
    for (int e = t; e < 2048; e += 64) {
        int rr = e >> 5, l = e & 31;
        float v;
        if (rr < 32) v = aRe[rr] * bRe[l] - aIm[rr] * bIm[l];
        else { int h = rr - 32; v = aRe[h] * bIm[l] + aIm[h] * bRe[l]; }
        Xbuf[0][rr * XS + l] = v;
    }

    // ---- entangler layers ----
    for (int layer = 0; layer < DEPTH; ++layer) {
        if (t < NQ) {
            float half = 0.5f * weights[layer * NQ + t];
            wc[t] = cosf(half);
            ws[t] = sinf(half);
        }
        __syncthreads();  // also covers Xbuf[0] init on layer 0

        // Build Mhi/Mlo.  Kron entry: K[r,c] = (-i)^popc(r^c) * prod((r^c)bit ? sin : cos).
        // CNOT cascade C(0,1)..C(3,4) folded as row permutation rp = prefix-xor(r) (MSB-first).
        for (int e = t; e < 2048; e += 64) {
            int side = e >> 10;                 // 0 = hi (wires 0..4), 1 = lo (wires 5..9)
            int rc = e & 1023;
            int r = rc >> 5, c = rc & 31;
            const float* cc2 = side ? (wc + 5) : wc;
            const float* ss2 = side ? (ws + 5) : ws;
            int x = r ^ c;
            float m = 1.0f;
            #pragma unroll
            for (int j = 0; j < 5; ++j)
                m *= ((x >> (4 - j)) & 1) ? ss2[j] : cc2[j];
            int k = __popc(x) & 3;
            float re = (k == 0) ? m : ((k == 2) ? -m : 0.0f);
            float im = (k == 3) ? m : ((k == 1) ? -m : 0.0f);
            int acc2 = 0, rp = 0;
            #pragma unroll
            for (int j = 0; j < 5; ++j) {
                acc2 ^= (r >> (4 - j)) & 1;
                rp |= acc2 << (4 - j);
            }
            float* M = side ? Mlo : Mhi;
            M[rp * MS + c]             =  re;   // [[Re, -Im],
            M[rp * MS + c + 32]        = -im;   //  [Im,  Re]]
            M[(rp + 32) * MS + c]      =  im;
            M[(rp + 32) * MS + c + 32] =  re;
        }
        __syncthreads();

        // pass 0: U_hi : Xbuf[0] -> Xbuf[1] (transposed)   then Cross45 on Xbuf[1]
        // pass 1: U_lo : Xbuf[1] -> Xbuf[0] (transposed)   then Cross90 on Xbuf[0]
        for (int pass = 0; pass < 2; ++pass) {
            const float* M    = pass ? Mlo : Mhi;
            const float* Xin  = Xbuf[pass ? 1 : 0];
            float*       Xout = Xbuf[pass ? 0 : 1];
            const int row   = lane & 15;    // M-row within tile / N within tile
            const int khalf = lane >> 4;    // K half select (ISA A/B lane layout)

            // C(64x32) = M(64x64) * Xin(64x32): 4x2 output tiles, 16 K-steps of 4.
            for (int tile = wave; tile < 8; tile += 2) {   // uniform within each wave
                int i = tile >> 1;          // row-tile 0..3
                int j = tile & 1;           // col-tile 0..1
                v8f acc = {0.f,0.f,0.f,0.f,0.f,0.f,0.f,0.f};
                const float* Abase = M + (i * 16 + row) * MS + khalf * 2;
                const float* Bbase = Xin + j * 16 + row;
                #pragma unroll 4
                for (int kk = 0; kk < 16; ++kk) {
                    v2f a, b;
                    // A 16x4: lane<16 -> K=0,1 ; lane>=16 -> K=2,3 (VGPR0/1)
                    a.x = Abase[kk * 4 + 0];
                    a.y = Abase[kk * 4 + 1];
                    // B 4x16: same K-half pattern, N = lane&15
                    int kb = kk * 4 + khalf * 2;
                    b.x = Bbase[(kb + 0) * XS];
                    b.y = Bbase[(kb + 1) * XS];
                    acc = __builtin_amdgcn_wmma_f32_16x16x4_f32(
                        /*neg_a=*/false, a, /*neg_b=*/false, b,
                        /*c_mod=*/(short)0, acc, /*reuse_a=*/false, /*reuse_b=*/false);
                }
                // C tile: VGPR v, lane<16 -> M=v ; lane>=16 -> M=v+8 ; N=lane&15.
                // Store complex-transposed: (gr<32: Re) -> Xout[gc][gr], (gr>=32: Im) -> Xout[32+gc][gr-32]
                int gc = j * 16 + (lane & 15);
                #pragma unroll
                for (int v = 0; v < 8; ++v) {
                    int gr = i * 16 + khalf * 8 + v;
                    int off = (gr < 32) ? (gc * XS + gr) : ((32 + gc) * XS + (gr - 32));
                    Xout[off] = acc[v];
                }
            }
            __syncthreads();

            // Cross CNOT (both orientations): swap complex rows r<->r+16 on odd cols
            // => real rows {0..15,32..47} <-> +16, odd cols.  512 swaps, one thread each.
            for (int e = t; e < 512; e += 64) {
                int rr = e >> 4;
                int r = (rr < 16) ? rr : (rr + 16);
                int c = ((e & 15) << 1) | 1;
                float u0 = Xout[r * XS + c];
                float u1 = Xout[(r + 16) * XS + c];
                Xout[r * XS + c]        = u1;
                Xout[(r + 16) * XS + c] = u0;
            }
            __syncthreads();
        }
    }

    // ---- p000 = sum over trash (wires 7..9 => l%8==0) of |S|^2 ; out = (1-p000)/2 ----
    {
        float s = 0.0f;
        #pragma unroll
        for (int c = 0; c < 32; c += 8) {
            float v = Xbuf[0][t * XS + c];
            s += v * v;
        }
        red[t] = s;
        __syncthreads();
        if (t == 0) {
            float p = 0.0f;
            for (int u = 0; u < 64; ++u) p += red[u];   // deterministic serial sum
            out[samp] = 0.5f * (1.0f - p);
        }
    }
}

extern "C" void kernel_launch(void* const* d_in, const int* in_sizes, int n_in,
                              void* d_out, int out_size, void* d_ws, size_t ws_size,
                              hipStream_t stream) {
    (void)in_sizes; (void)n_in; (void)d_ws; (void)ws_size; (void)out_size;
    const float* features = (const float*)d_in[0];   // [256][10] f32
    const float* weights  = (const float*)d_in[1];   // [4][10]   f32
    float* out = (float*)d_out;                      // [256]     f32
    qae_kernel<<<256, 64, 0, stream>>>(features, weights, out);
}